// SageConv_57973468561934
// MI455X (gfx1250) — compile-verified
//
#include <hip/hip_runtime.h>

#define N_NODES 100000
#define N_EDGES 1600000
#define D       128      // D_IN == D_OUT
#define D_OUT2  256      // concat(local, neigh)

typedef __attribute__((ext_vector_type(2))) float v2f;
typedef __attribute__((ext_vector_type(8))) float v8f;

// ---------------------------------------------------------------------------
// Zero the aggregation workspace (d_ws is poisoned with 0xAA by the harness).
// ---------------------------------------------------------------------------
__global__ void sage_zero(float4* __restrict__ p, int n4) {
    int i = blockIdx.x * blockDim.x + threadIdx.x;
    if (i < n4) p[i] = make_float4(0.f, 0.f, 0.f, 0.f);
}

// ---------------------------------------------------------------------------
// COO SpMM scatter: agg[row] += val * x[col].
// One wave (32 lanes) per edge; each lane owns a float4 chunk of the 128-wide
// feature row. x (51.2 MB) and agg (51.2 MB) are L2-resident on MI455X
// (192 MB L2), so the gather + fp32 atomics stay on-chip.
// ---------------------------------------------------------------------------
__global__ __launch_bounds__(256) void sage_spmm_scatter(
    const float* __restrict__ x,
    const int*   __restrict__ erow,
    const int*   __restrict__ ecol,
    const float* __restrict__ eval,
    float*       __restrict__ agg)
{
    int t    = blockIdx.x * blockDim.x + threadIdx.x;
    int e    = t >> 5;
    int lane = t & 31;
    if (e >= N_EDGES) return;

    int   r = erow[e];
    int   c = ecol[e];
    float v = eval[e];

    const float4 m = *((const float4*)(x + (size_t)c * D) + lane);
    float* dst = agg + (size_t)r * D + lane * 4;
    unsafeAtomicAdd(dst + 0, v * m.x);   // global_atomic_add_f32
    unsafeAtomicAdd(dst + 1, v * m.y);
    unsafeAtomicAdd(dst + 2, v * m.z);
    unsafeAtomicAdd(dst + 3, v * m.w);
}

// ---------------------------------------------------------------------------
// Fused dual GEMM + bias + ReLU + concat, using V_WMMA_F32_16X16X4_F32.
// Block = 256 threads = 8 waves; block owns a 16-node tile.
// Wave w computes output col-tile w for BOTH gemms (g=0: x@W_local,
// g=1: agg@W_neigh), each as a 16x16 f32 tile with a K=128 loop (32 wmma).
// ---------------------------------------------------------------------------
__global__ __launch_bounds__(256) void sage_gemm(
    const float* __restrict__ x,     // [N, 128]
    const float* __restrict__ agg,   // [N, 128]
    const float* __restrict__ Wl,    // [128, 128] (row = in, col = out)
    const float* __restrict__ bl,    // [128]
    const float* __restrict__ Wn,    // [128, 128]
    const float* __restrict__ bn,    // [128]
    float*       __restrict__ out)   // [N, 256]
{
    const int lane  = threadIdx.x & 31;
    const int wave  = threadIdx.x >> 5;          // 0..7
    const int node0 = blockIdx.x * 16;

    const int m   = lane & 15;                   // A row / D column within tile
    const int kb  = (lane >> 4) << 1;            // 0 or 2: K sub-offset per lane half
    const int mhi = (lane >> 4) << 3;            // +8 row offset for lanes 16..31

    #pragma unroll
    for (int t = wave; t < 16; t += 8) {
        const int g  = t >> 3;                   // 0 = local, 1 = neigh
        const int ct = t & 7;                    // 16-wide column tile within 128
        const float* __restrict__ A = g ? agg : x;
        const float* __restrict__ W = g ? Wn  : Wl;
        const float* __restrict__ b = g ? bn  : bl;
        const int n = ct * 16 + m;               // output column (0..127)

        const float* arow = A + (size_t)(node0 + m) * D;

        v8f acc = {};
        #pragma unroll
        for (int k = 0; k < D; k += 4) {
            // A 16x4: lane l -> A[m][k+kb], A[m][k+kb+1]  (contiguous b64 load)
            v2f av = *(const v2f*)(arow + k + kb);
            // B 4x16: lane l -> W[k+kb][n], W[k+kb+1][n]
            v2f bv;
            bv.x = W[(size_t)(k + kb)     * D + n];
            bv.y = W[(size_t)(k + kb + 1) * D + n];
            acc = __builtin_amdgcn_wmma_f32_16x16x4_f32(
                false, av, false, bv, (short)0, acc, false, false);
        }

        const float bias = b[n];
        #pragma unroll
        for (int r = 0; r < 8; ++r) {
            const int mm = r + mhi;              // row within the 16-node tile
            float v = acc[r] + bias;
            v = v > 0.f ? v : 0.f;               // ReLU
            out[(size_t)(node0 + mm) * D_OUT2 + g * D + ct * 16 + m] = v;
        }
    }
}

// ---------------------------------------------------------------------------
extern "C" void kernel_launch(void* const* d_in, const int* in_sizes, int n_in,
                              void* d_out, int out_size, void* d_ws, size_t ws_size,
                              hipStream_t stream) {
    const float* x    = (const float*)d_in[0];
    const int*   erow = (const int*)  d_in[1];
    const int*   ecol = (const int*)  d_in[2];
    const float* eval = (const float*)d_in[3];
    const float* Wl   = (const float*)d_in[4];
    const float* bl   = (const float*)d_in[5];
    const float* Wn   = (const float*)d_in[6];
    const float* bn   = (const float*)d_in[7];

    float* out = (float*)d_out;
    float* agg = (float*)d_ws;                   // [N_NODES, 128] fp32 scratch

    // 1) agg = 0
    const int n4 = (N_NODES * D) / 4;            // 3,200,000 float4
    sage_zero<<<(n4 + 255) / 256, 256, 0, stream>>>((float4*)agg, n4);

    // 2) agg[row] += val * x[col]   (one wave per edge)
    const long long spmm_threads = (long long)N_EDGES * 32;
    sage_spmm_scatter<<<(int)(spmm_threads / 256), 256, 0, stream>>>(
        x, erow, ecol, eval, agg);

    // 3) out = relu([x@Wl + bl, agg@Wn + bn])   (WMMA f32)
    sage_gemm<<<N_NODES / 16, 256, 0, stream>>>(x, agg, Wl, bl, Wn, bn, out);
}